// DetectionDecoder_89910845375157
// MI455X (gfx1250) — compile-verified
//
#include <hip/hip_runtime.h>
#include <hip/hip_bf16.h>
#include <stdint.h>

#define NBOX     16384
#define NCLASS   91          // input classes incl. background
#define NCLS     90          // classes 1..90 actually NMS'd
#define NBATCH   16
#define MAXSEL   100
#define TOPK     200
#define THREADS  256
#define PER_TH   (NBOX / THREADS)   // 64 scores per thread, kept in VGPRs
#define NEGV     (-1e30f)
#define NEGH     (-1e29f)
#define SCORE_TH 0.3f
#define IOU_TH   0.5f

// Number of boxes cached in LDS (16 B each). 16384 -> 256 KB (CDNA5 WGP has
// 320 KB per workgroup). Lower this if the toolchain enforces a smaller cap;
// boxes with index >= BOXC_CAP fall back to (L2-resident) global loads.
#define BOXC_CAP 16384

// ---------------------------------------------------------------------------
// Block-wide argmax with exact first-index tie-break (wave32 shuffles + LDS).
// ---------------------------------------------------------------------------
__device__ __forceinline__ void wave_argmax(float& bv, int& bi) {
#pragma unroll
  for (int off = 16; off > 0; off >>= 1) {
    float ov = __shfl_down(bv, off, 32);
    int   oi = __shfl_down(bi, off, 32);
    if (ov > bv || (ov == bv && oi < bi)) { bv = ov; bi = oi; }
  }
}

// CDNA5 async global->LDS copy: 16 bytes per active lane, tracked on ASYNCcnt.
__device__ __forceinline__ void async_b128_to_lds(void* lds_dst, const void* gsrc) {
  uint32_t loff = (uint32_t)(uintptr_t)lds_dst;
  asm volatile("global_load_async_to_lds_b128 %0, %1, off"
               :: "v"(loff), "v"(gsrc) : "memory");
}
__device__ __forceinline__ void wait_asynccnt0() {
  asm volatile("s_wait_asynccnt 0" ::: "memory");
}

// ---------------------------------------------------------------------------
// Kernel 1: exact greedy NMS, one workgroup per (class, batch).
// Scores live in registers (64/thread). The per-batch box array (256 KB) is
// staged once into LDS via the CDNA5 async global->LDS DMA path, so all 100
// suppression sweeps read boxes from LDS instead of L2.
// ---------------------------------------------------------------------------
__global__ __launch_bounds__(THREADS) void nms_kernel(
    const float* __restrict__ scores,   // [B, N, 91]
    const float* __restrict__ boxes,    // [B, N, 4]  (y1,x1,y2,x2)
    float* __restrict__ ws_scores,      // [B, NCLS, MAXSEL]
    float* __restrict__ ws_boxes)       // [B, NCLS, MAXSEL, 4]
{
  const int c    = blockIdx.x;          // 0..89 -> class id c+1
  const int b    = blockIdx.y;
  const int tid  = threadIdx.x;
  const int lane = tid & 31;
  const int wave = tid >> 5;

  __shared__ __align__(16) float4 boxc[BOXC_CAP];   // 256 KB box cache
  __shared__ float pv[8];
  __shared__ int   pi[8];
  __shared__ float bval;
  __shared__ int   bidx;

  const float* sc = scores + (size_t)b * NBOX * NCLASS + (c + 1);
  const float* bx = boxes  + (size_t)b * NBOX * 4;

  // ---- stage boxes into LDS with async DMA (overlaps the score loads) ----
#pragma unroll 4
  for (int k = 0; k < BOXC_CAP / THREADS; ++k) {
    const int i = k * THREADS + tid;
    async_b128_to_lds(&boxc[i], bx + (size_t)i * 4);
  }

  // Load + threshold this class's scores into registers (masked like ref).
  float arr[PER_TH];
#pragma unroll
  for (int k = 0; k < PER_TH; ++k) {
    float s = sc[(size_t)(k * THREADS + tid) * NCLASS];
    arr[k] = (s > SCORE_TH) ? s : NEGV;
  }

  wait_asynccnt0();      // box cache complete for this wave's transfers
  __syncthreads();       // ... and visible to all waves

  float* osc = ws_scores + (size_t)(b * NCLS + c) * MAXSEL;
  float* obx = ws_boxes  + (size_t)(b * NCLS + c) * MAXSEL * 4;

  int iter = 0;
  for (; iter < MAXSEL; ++iter) {
    // ---- block argmax (first-index tie-break, disjoint index sets) ----
    float bv = NEGV; int bi = 0x7fffffff;
#pragma unroll
    for (int k = 0; k < PER_TH; ++k) {
      if (arr[k] > bv) { bv = arr[k]; bi = k * THREADS + tid; }
    }
    wave_argmax(bv, bi);
    if (lane == 0) { pv[wave] = bv; pi[wave] = bi; }
    __syncthreads();

    if (tid == 0) {
      float v = pv[0]; int ix = pi[0];
#pragma unroll
      for (int w = 1; w < 8; ++w) {
        if (pv[w] > v || (pv[w] == v && pi[w] < ix)) { v = pv[w]; ix = pi[w]; }
      }
      bval = v; bidx = ix;
    }
    __syncthreads();

    const float v = bval;
    if (!(v > NEGH)) break;           // no candidates left: remaining slots 0
    const int six = bidx;

    // Selected box: uniform-address LDS read -> hardware broadcast.
    const float4 sb = (six < BOXC_CAP) ? boxc[six]
                                       : *(const float4*)(bx + (size_t)six * 4);
    const float sy1 = sb.x, sx1 = sb.y, sy2 = sb.z, sx2 = sb.w;

    if (tid == 0) {
      osc[iter] = v;
      obx[iter * 4 + 0] = sy1; obx[iter * 4 + 1] = sx1;
      obx[iter * 4 + 2] = sy2; obx[iter * 4 + 3] = sx2;
    }

    const float asel = (sy2 - sy1) * (sx2 - sx1);
    // ---- suppression sweep: IoU(selected, alive candidates), boxes in LDS ----
#pragma unroll
    for (int k = 0; k < PER_TH; ++k) {
      const int i = k * THREADS + tid;
      if (i == six) { arr[k] = NEGV; continue; }    // arange==idx term
      if (arr[k] <= NEGH) continue;                 // already masked
      const float4 bb = (i < BOXC_CAP) ? boxc[i]
                                       : *(const float4*)(bx + (size_t)i * 4);
      const float iy1 = fmaxf(sy1, bb.x), ix1 = fmaxf(sx1, bb.y);
      const float iy2 = fminf(sy2, bb.z), ix2 = fminf(sx2, bb.w);
      const float dy = iy2 - iy1, dx = ix2 - ix1;
      if (dy > 0.f && dx > 0.f) {
        const float inter = dy * dx;
        const float uni   = asel + (bb.z - bb.x) * (bb.w - bb.y) - inter;
        const float iou   = (uni > 0.f) ? (inter / uni) : 0.f;
        if (iou > IOU_TH) arr[k] = NEGV;
      }
    }
    // No extra barrier needed: pv/bval reuse is fenced by the two barriers above.
  }

  // Zero-fill unselected slots (matches reference padding semantics).
  for (int m = iter + tid; m < MAXSEL; m += THREADS) {
    osc[m] = 0.f;
    obx[m * 4 + 0] = 0.f; obx[m * 4 + 1] = 0.f;
    obx[m * 4 + 2] = 0.f; obx[m * 4 + 3] = 0.f;
  }
}

// ---------------------------------------------------------------------------
// Kernel 2: per-batch top-200 over 9000 slots, exact top_k order
// (descending score, ties -> lowest flat index). One workgroup per batch.
// ---------------------------------------------------------------------------
__global__ __launch_bounds__(THREADS) void topk_kernel(
    const float* __restrict__ ws_scores,  // [B, 9000]
    const float* __restrict__ ws_boxes,   // [B, 9000, 4]
    float* __restrict__ out)              // [B,200,2] scores+ids, then [B,200,4] boxes
{
  const int b    = blockIdx.x;
  const int tid  = threadIdx.x;
  const int lane = tid & 31;
  const int wave = tid >> 5;

  __shared__ float s[NCLS * MAXSEL];      // 9000 floats = 36 KB
  __shared__ float pv[8];
  __shared__ int   pi[8];

  const float* in_s = ws_scores + (size_t)b * (NCLS * MAXSEL);
  const float* in_b = ws_boxes  + (size_t)b * (NCLS * MAXSEL) * 4;
  float* out_s = out + (size_t)b * TOPK * 2;
  float* out_b = out + (size_t)NBATCH * TOPK * 2 + (size_t)b * TOPK * 4;

  for (int i = tid; i < NCLS * MAXSEL; i += THREADS) s[i] = in_s[i];
  __syncthreads();

  for (int k = 0; k < TOPK; ++k) {
    float bv = NEGV; int bi = 0x7fffffff;
    for (int i = tid; i < NCLS * MAXSEL; i += THREADS) {
      if (s[i] > bv) { bv = s[i]; bi = i; }
    }
    wave_argmax(bv, bi);
    if (lane == 0) { pv[wave] = bv; pi[wave] = bi; }
    __syncthreads();
    if (tid == 0) {
      float v = pv[0]; int ix = pi[0];
#pragma unroll
      for (int w = 1; w < 8; ++w) {
        if (pv[w] > v || (pv[w] == v && pi[w] < ix)) { v = pv[w]; ix = pi[w]; }
      }
      s[ix] = NEGV;                                   // consume this slot
      // valid selections always have score > 0.3 > 0; padded slots are 0.
      out_s[k * 2 + 0] = (v > 0.f) ? (float)(ix / MAXSEL + 1) : 0.f;
      out_s[k * 2 + 1] = v;
      const float4 bb = *(const float4*)(in_b + (size_t)ix * 4);
      *(float4*)(out_b + (size_t)k * 4) = bb;
    }
    __syncthreads();
  }
}

// ---------------------------------------------------------------------------
extern "C" void kernel_launch(void* const* d_in, const int* in_sizes, int n_in,
                              void* d_out, int out_size, void* d_ws, size_t ws_size,
                              hipStream_t stream) {
  (void)in_sizes; (void)n_in; (void)out_size; (void)ws_size;
  const float* scores = (const float*)d_in[0];   // [16,16384,91] f32
  const float* boxes  = (const float*)d_in[1];   // [16,16384,4]  f32

  float* ws_scores = (float*)d_ws;                                  // 144000 f
  float* ws_boxes  = ws_scores + (size_t)NBATCH * NCLS * MAXSEL;    // 576000 f

  dim3 g1(NCLS, NBATCH);
  nms_kernel<<<g1, THREADS, 0, stream>>>(scores, boxes, ws_scores, ws_boxes);
  topk_kernel<<<NBATCH, THREADS, 0, stream>>>(ws_scores, ws_boxes, (float*)d_out);
}